// DigitConvolutionalModel_65953517797796
// MI455X (gfx1250) — compile-verified
//
#include <hip/hip_runtime.h>

// CDNA5 / gfx1250 fused kernel: conv3x3(ones) + GEMM(676->128,relu) + GEMM(128->10)
// GEMM1 uses V_WMMA_F32_16X16X4_F32 (exact fp32 — matches fp32 reference).

typedef float v2f __attribute__((ext_vector_type(2)));
typedef float v8f __attribute__((ext_vector_type(8)));

#define THREADS   256
#define ROWS      32          // batch rows per workgroup (two 16-row M tiles)
#define HW        784         // 28*28
#define IMG_W     28
#define CONV_W    26
#define FLAT      676         // 26*26
#define HID       128
#define OUT_N     10

__global__ __launch_bounds__(THREADS)
void digit_model_fused(const float* __restrict__ x,
                       const float* __restrict__ conv_w,
                       const float* __restrict__ w1,
                       const float* __restrict__ b1,
                       const float* __restrict__ w2,
                       const float* __restrict__ b2,
                       float* __restrict__ out)
{
    extern __shared__ float smem[];
    float* xs   = smem;                       // ROWS * 784 floats (100 KB)
    float* hbuf = xs + ROWS * HW;             // ROWS * 676 floats (86.5 KB), pitch 676 (bank-conflict-free)
    float* h2   = hbuf + ROWS * FLAT;         // ROWS * 128 floats (16 KB)

    const int tid  = threadIdx.x;
    const int row0 = blockIdx.x * ROWS;

    // ---------------- stage x tile to LDS (coalesced float4) ----------------
    {
        const float4* src = reinterpret_cast<const float4*>(x + (size_t)row0 * HW);
        float4*       dst = reinterpret_cast<float4*>(xs);
        for (int i = tid; i < ROWS * HW / 4; i += THREADS)
            dst[i] = src[i];
    }

    // 3x3 conv weights (broadcast scalar loads)
    float cw[9];
#pragma unroll
    for (int i = 0; i < 9; ++i) cw[i] = conv_w[i];

    __syncthreads();

    // ---------------- 3x3 valid conv: xs(32x28x28) -> hbuf(32x676) ----------------
    for (int idx = tid; idx < ROWS * FLAT; idx += THREADS) {
        int r = idx / FLAT;
        int p = idx - r * FLAT;
        int i = p / CONV_W;
        int j = p - i * CONV_W;
        const float* base = &xs[r * HW + i * IMG_W + j];
        float acc = 0.f;
#pragma unroll
        for (int a = 0; a < 3; ++a)
#pragma unroll
            for (int b = 0; b < 3; ++b)
                acc = fmaf(base[a * IMG_W + b], cw[a * 3 + b], acc);
        hbuf[r * FLAT + p] = acc;
    }
    __syncthreads();

    // ---------------- GEMM1: h(32x676) @ w1^T(676x128) via V_WMMA_F32_16X16X4_F32 ----
    // Wave w owns hidden tile n in [16w, 16w+16); computes two M tiles (rows 0-15, 16-31).
    // f32 16x16x4 fragment layout (ISA 7.12.2):
    //   A: lane L, vgpr j -> A[L&15][2*(L>>4) + j]
    //   B: lane L, vgpr j -> B[2*(L>>4) + j][L&15]
    //   C: lane L, vgpr r -> C[r + 8*(L>>4)][L&15]
    const int lane = tid & 31;
    const int wav  = tid >> 5;      // 0..7
    const int hh   = lane >> 4;     // 0/1 (half-wave)
    const int nl   = lane & 15;
    const int n    = wav * 16 + nl; // hidden unit column for this lane

    v8f c0 = {};
    v8f c1 = {};
    const float* aRow0 = &hbuf[(nl)      * FLAT + 2 * hh]; // LDS, M tile 0
    const float* aRow1 = &hbuf[(16 + nl) * FLAT + 2 * hh]; // LDS, M tile 1
    const float* bRow  = &w1[(size_t)n * FLAT + 2 * hh];   // global (L2-resident, 346 KB)

#pragma unroll 4
    for (int k = 0; k < FLAT; k += 4) {
        v2f b = *reinterpret_cast<const v2f*>(bRow + k);   // global_load_b64
        v2f a0 = *reinterpret_cast<const v2f*>(aRow0 + k); // ds_load_b64
        v2f a1 = *reinterpret_cast<const v2f*>(aRow1 + k); // ds_load_b64
        c0 = __builtin_amdgcn_wmma_f32_16x16x4_f32(false, a0, false, b, (short)0, c0, false, false);
        c1 = __builtin_amdgcn_wmma_f32_16x16x4_f32(false, a1, false, b, (short)0, c1, false, false);
    }

    // bias + ReLU, scatter to h2 LDS: h2[m][n]
    const float bias = b1[n];
#pragma unroll
    for (int r = 0; r < 8; ++r) {
        int m = r + 8 * hh;
        float v0 = c0[r] + bias;
        float v1 = c1[r] + bias;
        h2[(m)      * HID + n] = v0 > 0.f ? v0 : 0.f;
        h2[(m + 16) * HID + n] = v1 > 0.f ? v1 : 0.f;
    }
    __syncthreads();

    // ---------------- GEMM2: h2(32x128) @ w2^T(128x10) + b2 ----------------
    for (int o = tid; o < ROWS * OUT_N; o += THREADS) {
        int m = o / OUT_N;
        int q = o - m * OUT_N;
        const float* hv = &h2[m * HID];
        const float* wv = &w2[q * HID];
        float acc = b2[q];
#pragma unroll 8
        for (int k = 0; k < HID; ++k)
            acc = fmaf(hv[k], wv[k], acc);
        out[(size_t)(row0 + m) * OUT_N + q] = acc;
    }
}

extern "C" void kernel_launch(void* const* d_in, const int* in_sizes, int n_in,
                              void* d_out, int out_size, void* d_ws, size_t ws_size,
                              hipStream_t stream) {
    const float* x      = (const float*)d_in[0];
    const float* conv_w = (const float*)d_in[1];
    const float* w1     = (const float*)d_in[2];
    const float* b1     = (const float*)d_in[3];
    const float* w2     = (const float*)d_in[4];
    const float* b2     = (const float*)d_in[5];
    float* out          = (float*)d_out;

    const int B = in_sizes[0] / HW;                 // 65536
    const int grid = B / ROWS;                      // 2048 workgroups
    const size_t shmem = (size_t)(ROWS * HW + ROWS * FLAT + ROWS * HID) * sizeof(float); // ~203 KB

    digit_model_fused<<<dim3(grid), dim3(THREADS), shmem, stream>>>(
        x, conv_w, w1, b1, w2, b2, out);
}